// LocalSingularityStrength_52493090291861
// MI455X (gfx1250) — compile-verified
//
#include <hip/hip_runtime.h>

#define H_DIM 224
#define W_DIM 224
#define C_DIM 32
#define B_DIM 16
#define TW 16         // output cols per tile
#define TH 16         // output rows per tile
#define RROWS 31      // TH + 15 halo rows
#define RCOLS 31      // TW + 15 halo cols
#define ROWFL (RCOLS * C_DIM)   // 992 floats per staged row
#define ROWV4 (ROWFL / 4)       // 248 x b128 chunks per row
#define NW (TW * C_DIM)         // 512 (w,c) pairs per tile row
#define RS2 1056      // pair-row stride (2*512 data + 32 pad; 1056%64==32)
#define PLANE (16 * RS2)        // floats per h-plane (16 row-pairs)
#define EPSF 1e-7f
#define BN_EPSF 1e-3f

typedef __attribute__((ext_vector_type(2))) float v2f;
typedef __attribute__((ext_vector_type(8))) float v8f;

// Monotone order-preserving float<->uint key (for atomic min/max).
__device__ __forceinline__ unsigned fkey(float f) {
  unsigned u = __float_as_uint(f);
  return (u & 0x80000000u) ? ~u : (u | 0x80000000u);
}
__device__ __forceinline__ float funkey(unsigned e) {
  unsigned u = (e & 0x80000000u) ? (e & 0x7fffffffu) : ~e;
  return __uint_as_float(u);
}

__global__ void lss_init_keys(unsigned* keys) {
  int t = threadIdx.x;
  if (t < B_DIM) {
    keys[t] = 0xFFFFFFFFu;       // min keys
    keys[B_DIM + t] = 0u;        // max keys
  }
}

// Pass 1: per-sample min/max over H*W*C. float4 loads, wave32 xor-reduce,
// one atomic pair per wave.
__global__ void lss_minmax(const float* __restrict__ x,
                           unsigned* __restrict__ keys) {
  const int b = blockIdx.y;
  const float4* xb = (const float4*)(x + (size_t)b * H_DIM * W_DIM * C_DIM);
  const int n4 = H_DIM * W_DIM * C_DIM / 4;
  unsigned kmin = 0xFFFFFFFFu, kmax = 0u;
  for (int i = blockIdx.x * blockDim.x + threadIdx.x; i < n4;
       i += gridDim.x * blockDim.x) {
    float4 v = xb[i];
    unsigned a = fkey(v.x), bb = fkey(v.y), c = fkey(v.z), d = fkey(v.w);
    kmin = min(kmin, min(min(a, bb), min(c, d)));
    kmax = max(kmax, max(max(a, bb), max(c, d)));
  }
#pragma unroll
  for (int off = 16; off > 0; off >>= 1) {
    kmin = min(kmin, (unsigned)__shfl_xor((int)kmin, off, 32));
    kmax = max(kmax, (unsigned)__shfl_xor((int)kmax, off, 32));
  }
  if ((threadIdx.x & 31) == 0) {
    atomicMin(&keys[b], kmin);
    atomicMax(&keys[B_DIM + b], kmax);
  }
}

// Pass 2: fused pipeline.
//  - rows staged raw into double-buffered LDS via async global->LDS b128 DMA
//    (ASYNCcnt-pipelined); borders padded with `mn` so the affine rescale
//    maps padding to exactly 0 (box sums are linear).
//  - hierarchical horizontal box sums in VALU (lane = channel), stored to
//    row-pair-interleaved h-planes: hb[s][j>>1][2*n + (j&1)] so each WMMA
//    B-fragment is a single ds_load_b64.
//  - vertical box sums = banded ones-matrix x h-planes via chained
//    V_WMMA_F32_16X16X4_F32; A-fragments extracted from per-scale 32-bit
//    window masks (bfe+cvt), precomputed once and reused per N-group.
//  - finalize: S_xs = inv*S_raw - inv*mn*r^2, log-slope OLS, BatchNorm.
__global__ void __launch_bounds__(256, 1)
lss_main(const float* __restrict__ x, const float* __restrict__ gamma,
         const float* __restrict__ beta, const float* __restrict__ mmean,
         const float* __restrict__ mvar, const unsigned* __restrict__ keys,
         float* __restrict__ out) {
  __shared__ __align__(16) float rowbuf[2 * ROWFL];   //  7.9 KB double buffer
  __shared__ __align__(16) float hb[4 * PLANE];       //  270 KB h2,h4,h8,h16

  const int t = threadIdx.x;
  const int lane = t & 31;
  const int wv = t >> 5;            // wave id 0..7
  const int b = blockIdx.z;
  const int h0 = blockIdx.y * TH;
  const int w0 = blockIdx.x * TW;

  const float mn = funkey(keys[b]);
  const float mx = funkey(keys[B_DIM + b]);
  const float inv = 1.0f / (mx - mn + EPSF);
  const size_t imgoff = (size_t)b * H_DIM * W_DIM * C_DIM;
  const bool colsfull = (w0 >= 7) && (w0 + TW + 8 <= W_DIM);  // full halo row

  // Zero the unused K-row (j=31, odd slot of pair-row 15): multiplied by an
  // A-matrix zero in the last WMMA chunk, must not contain NaN bits.
  for (int s = 0; s < 4; ++s)
    for (int e = t; e < NW; e += 256)
      hb[s * PLANE + 15 * RS2 + 2 * e + 1] = 0.0f;

  // Issue the staged load of halo row `ir` into rowbuf[ir&1].
  auto issue_row = [&](int ir) {
    const int gh = h0 - 7 + ir;
    float* buf = &rowbuf[(ir & 1) * ROWFL];
    const bool rowvalid = (unsigned)gh < (unsigned)H_DIM;
    if (rowvalid && colsfull) {
      // Async DMA: 248 x global_load_async_to_lds_b128, raw data.
      if (t < ROWV4) {
        const unsigned ldsoff = (unsigned)(size_t)buf + 16u * (unsigned)t;
        const unsigned voff = 16u * (unsigned)t;
        const float* rowp =
            x + imgoff + ((size_t)gh * W_DIM + (w0 - 7)) * C_DIM;
        asm volatile("global_load_async_to_lds_b128 %0, %1, %2"
                     :
                     : "v"(ldsoff), "v"(voff), "s"(rowp)
                     : "memory");
      }
    } else {
      // Border fallback: pad with mn so (pad - mn)*inv == 0.
      for (int e = t; e < ROWFL; e += 256) {
        const int j = e >> 5;
        const int c = e & 31;
        const int gw = w0 - 7 + j;
        float v = mn;
        if (rowvalid && (unsigned)gw < (unsigned)W_DIM)
          v = x[imgoff + ((size_t)gh * W_DIM + gw) * C_DIM + c];
        buf[e] = v;
      }
    }
  };

  const int cl = wv;     // wave owns output cols cl and cl+8
  const int ch = lane;   // channel = lane

  issue_row(0);
  for (int ir = 0; ir < RROWS; ++ir) {
    if (ir + 1 < RROWS) {
      issue_row(ir + 1);  // pipeline: next row in flight during compute
      // oldest async op (row ir) must have landed; in-order completion.
      asm volatile("s_wait_asynccnt 0x1" ::: "memory");
    } else {
      asm volatile("s_wait_asynccnt 0x0" ::: "memory");
    }
    __syncthreads();

    const float* buf = &rowbuf[(ir & 1) * ROWFL];
    // Shared 24-col window covers both output cols (cl+7 and cl+15).
    float xw[24];
#pragma unroll
    for (int k = 0; k < 24; ++k) xw[k] = buf[(cl + k) * C_DIM + ch];

    const int pr = (ir >> 1) * RS2 + (ir & 1);   // pair-row base + parity
#pragma unroll
    for (int ci = 0; ci < 2; ++ci) {
      const float* xv = &xw[ci * 8];
      const float h2 = xv[7] + xv[8];                              // [L,  L+1]
      const float h4 = h2 + (xv[6] + xv[9]);                       // [L-1,L+2]
      const float h8 = h4 + ((xv[4] + xv[5]) + (xv[10] + xv[11])); // [L-3,L+4]
      const float h16 = h8 + (((xv[0] + xv[1]) + (xv[2] + xv[3])) +
                              ((xv[12] + xv[13]) + (xv[14] + xv[15])));
      const int n = (cl + ci * 8) * C_DIM + ch;
      hb[0 * PLANE + pr + 2 * n] = h2;
      hb[1 * PLANE + pr + 2 * n] = h4;
      hb[2 * PLANE + pr + 2 * n] = h8;
      hb[3 * PLANE + pr + 2 * n] = h16;
    }
    __syncthreads();  // protect rowbuf[ir&1] before next issue overwrites it
  }

  // Vertical phase: S_r = A_r(16x31 band of ones) * h_r(31 x 512) via
  // chained V_WMMA_F32_16X16X4_F32, K tiled by 4 over halo rows.
  const int half = lane >> 4;     // 0: lanes 0-15, 1: lanes 16-31
  const int m = lane & 15;        // M (output row) for A / N index for B
  const int kk0 = half * 2;       // per-lane K slots {0,1} or {2,3}

  const int c0s[4] = {1, 1, 1, 0};   // first K-chunk touching each band
  const int c1s[4] = {5, 6, 6, 7};   // last  K-chunk touching each band
  const int pbs[4] = {0, 1, 3, 7};   // pad-before per scale (SAME)
  const int rr[4] = {2, 4, 8, 16};   // window length per scale

  // Per-scale window bitmask over halo rows j in [0,32):
  // bits [m+7-pb, m+7-pb+r) set.  A values come out via bfe+cvt.
  unsigned wmask[4];
#pragma unroll
  for (int s = 0; s < 4; ++s)
    wmask[s] = ((1u << rr[s]) - 1u) << (unsigned)(m + 7 - pbs[s]);

  // Precompute all 25 A-fragments once (depend only on lane/scale/chunk);
  // reused for every N-group.
  v2f amt[25];
  {
    int ai = 0;
#pragma unroll
    for (int s = 0; s < 4; ++s) {
#pragma unroll
      for (int q = c0s[s]; q <= c1s[s]; ++q) {
        const int j0 = q * 4 + kk0;
        v2f a;
        a.x = (float)((wmask[s] >> j0) & 1u);
        a.y = (float)((wmask[s] >> (j0 + 1)) & 1u);
        amt[ai++] = a;
      }
    }
  }

  // Affine fold: log2 argument = inv*S_raw + (EPS - inv*mn*r^2)
  const float ofs[4] = {EPSF - inv * mn * 4.0f, EPSF - inv * mn * 16.0f,
                        EPSF - inv * mn * 64.0f, EPSF - inv * mn * 256.0f};

  for (int gi = 0; gi < 4; ++gi) {
    const int g = wv + gi * 8;       // N-group: 16 (col,channel) pairs
    const int nidx = g * 16 + m;
    v8f accs[4];
    int ai = 0;
#pragma unroll
    for (int s = 0; s < 4; ++s) {
      v8f acc = {};
      const float* hs = &hb[s * PLANE];
#pragma unroll
      for (int q = c0s[s]; q <= c1s[s]; ++q) {
        // B-fragment: rows (j0, j0+1) are adjacent in pair-interleaved
        // layout -> single ds_load_b64.  pair-row = 2q + half.
        const v2f bm = *(const v2f*)&hs[(2 * q + half) * RS2 + 2 * nidx];
        acc = __builtin_amdgcn_wmma_f32_16x16x4_f32(
            false, amt[ai++], false, bm, (short)0, acc, false, false);
      }
      accs[s] = acc;
    }

    // All four scales for element (M,N) live in the same lane: finalize.
    const int gw = w0 + (g >> 1);
    const int c = (g & 1) * 16 + m;
    const float gm = gamma[c], bt = beta[c];
    const float mu = mmean[c];
    const float bninv = rsqrtf(mvar[c] + BN_EPSF);
#pragma unroll
    for (int v = 0; v < 8; ++v) {
      const int gh = h0 + v + half * 8;  // D layout: M = v + 8*half
      const float l2 = __log2f(fmaf(inv, accs[0][v], ofs[0]));
      const float l4 = __log2f(fmaf(inv, accs[1][v], ofs[1]));
      const float l8 = __log2f(fmaf(inv, accs[2][v], ofs[2]));
      const float l16 = __log2f(fmaf(inv, accs[3][v], ofs[3]));
      // OLS slope vs ln(scale): 0.2*(1.5*(log2 S16 - log2 S2)
      //                              + 0.5*(log2 S8 - log2 S4))
      const float alpha = 0.2f * (1.5f * (l16 - l2) + 0.5f * (l8 - l4));
      out[imgoff + ((size_t)gh * W_DIM + gw) * C_DIM + c] =
          (alpha - mu) * bninv * gm + bt;
    }
  }
}

extern "C" void kernel_launch(void* const* d_in, const int* in_sizes, int n_in,
                              void* d_out, int out_size, void* d_ws,
                              size_t ws_size, hipStream_t stream) {
  (void)in_sizes; (void)n_in; (void)out_size; (void)ws_size;
  const float* x = (const float*)d_in[0];
  const float* gamma = (const float*)d_in[1];
  const float* beta = (const float*)d_in[2];
  const float* mmean = (const float*)d_in[3];
  const float* mvar = (const float*)d_in[4];
  float* out = (float*)d_out;
  unsigned* keys = (unsigned*)d_ws;   // [2*B] uint keys, re-inited every call

  lss_init_keys<<<1, 32, 0, stream>>>(keys);
  lss_minmax<<<dim3(128, B_DIM), 256, 0, stream>>>(x, keys);
  lss_main<<<dim3(W_DIM / TW, H_DIM / TH, B_DIM), 256, 0, stream>>>(
      x, gamma, beta, mmean, mvar, keys, out);
}